// IDMPNN_Global_1065151889810
// MI455X (gfx1250) — compile-verified
//
#include <hip/hip_runtime.h>

// Problem constants (match reference)
#define B_      64
#define NM_     32
#define NS_     32
#define K_      3
#define HID_    64
#define NLAYER_ 4
#define P_      6
#define S_      (B_ * NS_)      // 2048 subgraphs
#define ROWS_   (P_ * NM_)      // 192 rows (perm-major: row = p*32 + n)
#define STR_    72              // padded LDS row stride (floats) -> conflict-free WMMA frag loads
#define ASTR_   33              // padded stride for 32x32 adjacency tile (TDM pad: 32+1)
#define WSTR_   72              // padded stride for 64x64 weight tiles   (TDM pad: 64+8)

typedef __attribute__((ext_vector_type(2))) float        v2f;
typedef __attribute__((ext_vector_type(8))) float        v8f;
typedef __attribute__((ext_vector_type(4))) unsigned int v4u;
typedef __attribute__((ext_vector_type(4))) int          v4i;
typedef __attribute__((ext_vector_type(8))) int          v8i;

// allperm[k][p] = p-th permutation of (0,1,2), element k (itertools order, transposed)
__constant__ int c_allperm[K_][P_] = {
  {0, 0, 1, 1, 2, 2},
  {1, 2, 0, 2, 0, 1},
  {2, 1, 2, 0, 1, 0},
};

__device__ __forceinline__ v8f v8f_zero() {
  v8f z = {0.f, 0.f, 0.f, 0.f, 0.f, 0.f, 0.f, 0.f};
  return z;
}

// D(16x16,f32) = A(16x4,f32) * B(4x16,f32) + C   -- CDNA5 dense f32 WMMA
__device__ __forceinline__ v8f wmma4(v2f a, v2f b, v8f c) {
  return __builtin_amdgcn_wmma_f32_16x16x4_f32(false, a, false, b, (short)0, c,
                                               false, false);
}

// ---- Tensor Data Mover: 2-D f32 tile (global -> LDS) with hardware row padding ----
// D# per CDNA5 ISA. Groups 2/3 zero (<=2D tensor).
// pad_interval: 0=2,1=4,...,4=32,5=64 DWORDs between pads; pad_amount: N+1 DWORDs.
__device__ __forceinline__ void tdm_load_2d_f32(unsigned lds_byte_off,
                                                const float* gptr,
                                                unsigned dim0, unsigned dim1,
                                                unsigned pad_interval,
                                                unsigned pad_amount) {
  unsigned long long ga = (unsigned long long)(uintptr_t)gptr;
  v4u g0;
  g0[0] = 1u;                                              // count=1, user mode
  g0[1] = lds_byte_off;                                    // lds_addr (bytes)
  g0[2] = (unsigned)(ga & 0xffffffffull);                  // global_addr[31:0]
  g0[3] = (unsigned)((ga >> 32) & 0x01ffffffull) | (2u << 30);  // addr[56:32] | type=2
  v8i g1;
  g1[0] = (int)((2u << 16) |                               // data_size = 4 bytes
                (1u << 20) |                               // pad_enable
                (pad_interval << 22) | (pad_amount << 25));
  g1[1] = (int)(dim0 << 16);                               // tensor_dim0[15:0] @ bit48
  g1[2] = (int)((dim0 >> 16) | (dim1 << 16));              // tensor_dim0 hi | tensor_dim1 lo
  g1[3] = (int)((dim1 >> 16) | (dim0 << 16));              // tensor_dim1 hi | tile_dim0
  g1[4] = (int)(dim1 & 0xffffu);                           // tile_dim1 (tile_dim2 = 0)
  g1[5] = (int)dim0;                                       // tensor_dim0_stride[31:0]
  g1[6] = 0;                                               // stride0 hi | stride1 lo
  g1[7] = 0;                                               // stride1 hi
  v4i gz = {0, 0, 0, 0};
#if defined(__clang_major__) && (__clang_major__ >= 23)
  v8i gz8 = {0, 0, 0, 0, 0, 0, 0, 0};
  __builtin_amdgcn_tensor_load_to_lds(g0, g1, gz, gz, gz8, 0);
#else
  __builtin_amdgcn_tensor_load_to_lds(g0, g1, gz, gz, 0);
#endif
}

__device__ __forceinline__ unsigned lds_off_of(const void* p) {
  return (unsigned)(unsigned long long)(uintptr_t)p;  // low 32 bits of flat LDS addr = byte offset
}

__global__ void init_pooled_kernel(unsigned int* __restrict__ pooled) {
  int i = blockIdx.x * blockDim.x + threadIdx.x;
  if (i < B_ * HID_) pooled[i] = 0u;  // bit pattern of +0.0f (all h2 outputs are >= 0)
}

__global__ __launch_bounds__(256)
void idmpnn_main_kernel(const int*   __restrict__ x,        // (B,NM,1)
                        const float* __restrict__ subadj,   // (B,NM,NM)
                        const int*   __restrict__ subgs,    // (S,K)
                        const int*   __restrict__ num_node, // (B,)
                        const float* __restrict__ idemb,    // (K+2,HID)
                        const float* __restrict__ node_emb, // (MAXZ+1,HID)
                        const float* __restrict__ W1, const float* __restrict__ B1,
                        const float* __restrict__ W2, const float* __restrict__ B2,
                        const float* __restrict__ G,  const float* __restrict__ BE,
                        const float* __restrict__ s1W, const float* __restrict__ s1b,
                        const float* __restrict__ s2W, const float* __restrict__ s2b,
                        unsigned int* __restrict__ pooled)  // (B,HID) as uint bits
{
  __shared__ float Ab[NM_ * ASTR_];     // adjacency 32x32, TDM-padded stride 33
  __shared__ float Wa[HID_ * WSTR_];    // 64x64 weight tile, TDM-padded stride 72
  __shared__ float Wb[HID_ * WSTR_];    // 64x64 weight tile, TDM-padded stride 72
  __shared__ float hb[ROWS_ * STR_];    // h : 192 x 64 (padded)
  __shared__ float mb[ROWS_ * STR_];    // m : scratch
  __shared__ float tb[ROWS_ * STR_];    // t : scratch
  __shared__ float s_mu[ROWS_];
  __shared__ float s_rs[ROWS_];
  __shared__ float s_col[HID_];

  const int s    = blockIdx.x;          // subgraph id
  const int b    = s / NS_;             // graph id
  const int tid  = threadIdx.x;
  const int lane = tid & 31;
  const int wave = tid >> 5;            // 8 waves
  const int half = lane >> 4;           // 0/1: lane group (K/M split per ISA layout)
  const int lr   = lane & 15;

  // ---- TDM: DMA adjacency tile into LDS (overlaps with h-build below) ----
  // 32x32 f32, pad 1 DWORD per 32 DWORDs -> LDS stride 33.
  if (wave == 0)
    tdm_load_2d_f32(lds_off_of(Ab), subadj + (size_t)b * NM_ * NM_,
                    NM_, NM_, /*interval 32dw*/4, /*pad 1dw*/0);

  const int sg0 = subgs[s * K_ + 0];
  const int sg1 = subgs[s * K_ + 1];
  const int sg2 = subgs[s * K_ + 2];

  // ---- build h[p][n][d] = node_emb[x[b,n]] (* idemb[perm_p[k]] if n is k-th subgraph node) ----
  for (int e = tid; e < ROWS_ * HID_; e += 256) {
    int row = e >> 6, d = e & 63;
    int p = row >> 5, n = row & 31;
    float v = node_emb[x[b * NM_ + n] * HID_ + d];
    int k = (n == sg0) ? 0 : (n == sg1) ? 1 : (n == sg2) ? 2 : -1;
    if (k >= 0) v *= idemb[c_allperm[k][p] * HID_ + d];
    hb[row * STR_ + d] = v;
  }
  if (wave == 0) __builtin_amdgcn_s_wait_tensorcnt(0);
  __syncthreads();

  // ---- message-passing layers ----
  for (int l = 0; l < NLAYER_; ++l) {
    const float* b1 = B1 + l * HID_;
    const float* b2 = B2 + l * HID_;
    const float* g  = G  + l * HID_;
    const float* be = BE + l * HID_;

    // TDM: DMA this layer's W1,W2 (64x64 f32, pad 8 DWORDs per 64 -> stride 72),
    // overlapping with the einsum which only touches Ab/hb.
    if (wave == 0) {
      tdm_load_2d_f32(lds_off_of(Wa), W1 + l * HID_ * HID_,
                      HID_, HID_, /*interval 64dw*/5, /*pad 8dw*/7);
      tdm_load_2d_f32(lds_off_of(Wb), W2 + l * HID_ * HID_,
                      HID_, HID_, /*interval 64dw*/5, /*pad 8dw*/7);
    }

    // m[p] = A @ h[p]   (per perm: 32x32 @ 32x64). 48 tiles / 8 waves (uniform EXEC).
    for (int t = wave; t < 48; t += 8) {
      int p = t >> 3, ti = (t >> 2) & 1, tj = t & 3;
      v8f acc = v8f_zero();
#pragma unroll
      for (int kk = 0; kk < NM_; kk += 4) {
        v2f a, bf;
        a.x  = Ab[(ti * 16 + lr) * ASTR_ + kk + 2 * half];
        a.y  = Ab[(ti * 16 + lr) * ASTR_ + kk + 2 * half + 1];
        bf.x = hb[(p * 32 + kk + 2 * half)     * STR_ + tj * 16 + lr];
        bf.y = hb[(p * 32 + kk + 2 * half + 1) * STR_ + tj * 16 + lr];
        acc = wmma4(a, bf, acc);
      }
      int rbase = p * 32 + ti * 16 + 8 * half;
      int c = tj * 16 + lr;
#pragma unroll
      for (int r = 0; r < 8; ++r) mb[(rbase + r) * STR_ + c] = acc[r];
    }
    if (wave == 0) __builtin_amdgcn_s_wait_tensorcnt(0);
    __syncthreads();

    // t = relu(m @ W1 + b1)   (192x64 @ 64x64), W1 fragments from LDS
    for (int t = wave; t < 48; t += 8) {
      int tr = t >> 2, tc = t & 3;
      v8f acc = v8f_zero();
#pragma unroll
      for (int kk = 0; kk < HID_; kk += 4) {
        v2f a, bf;
        a.x  = mb[(tr * 16 + lr) * STR_ + kk + 2 * half];
        a.y  = mb[(tr * 16 + lr) * STR_ + kk + 2 * half + 1];
        bf.x = Wa[(kk + 2 * half)     * WSTR_ + tc * 16 + lr];
        bf.y = Wa[(kk + 2 * half + 1) * WSTR_ + tc * 16 + lr];
        acc = wmma4(a, bf, acc);
      }
      int c = tc * 16 + lr;
      float bias = b1[c];
#pragma unroll
      for (int r = 0; r < 8; ++r) {
        float v = acc[r] + bias;
        tb[(tr * 16 + 8 * half + r) * STR_ + c] = v > 0.f ? v : 0.f;
      }
    }
    __syncthreads();

    // m = t @ W2 + b2 , W2 fragments from LDS
    for (int t = wave; t < 48; t += 8) {
      int tr = t >> 2, tc = t & 3;
      v8f acc = v8f_zero();
#pragma unroll
      for (int kk = 0; kk < HID_; kk += 4) {
        v2f a, bf;
        a.x  = tb[(tr * 16 + lr) * STR_ + kk + 2 * half];
        a.y  = tb[(tr * 16 + lr) * STR_ + kk + 2 * half + 1];
        bf.x = Wb[(kk + 2 * half)     * WSTR_ + tc * 16 + lr];
        bf.y = Wb[(kk + 2 * half + 1) * WSTR_ + tc * 16 + lr];
        acc = wmma4(a, bf, acc);
      }
      int c = tc * 16 + lr;
      float bias = b2[c];
#pragma unroll
      for (int r = 0; r < 8; ++r)
        mb[(tr * 16 + 8 * half + r) * STR_ + c] = acc[r] + bias;
    }
    __syncthreads();

    // LayerNorm stats per row
    if (tid < ROWS_) {
      float mu = 0.f;
      for (int d = 0; d < HID_; ++d) mu += mb[tid * STR_ + d];
      mu *= (1.f / 64.f);
      float var = 0.f;
      for (int d = 0; d < HID_; ++d) {
        float df = mb[tid * STR_ + d] - mu;
        var += df * df;
      }
      var *= (1.f / 64.f);
      s_mu[tid] = mu;
      s_rs[tid] = rsqrtf(var + 1e-5f);
    }
    __syncthreads();

    // h += relu(layernorm(m) * g + be)
    for (int e = tid; e < ROWS_ * HID_; e += 256) {
      int row = e >> 6, d = e & 63;
      float v = (mb[row * STR_ + d] - s_mu[row]) * s_rs[row] * g[d] + be[d];
      hb[row * STR_ + d] += v > 0.f ? v : 0.f;
    }
    __syncthreads();
  }

  // ---- post: TDM set1_W while computing the masked perm-mean ----
  if (wave == 0)
    tdm_load_2d_f32(lds_off_of(Wa), s1W, HID_, HID_, 5, 7);

  const int nn = num_node[b];
  for (int e = tid; e < NM_ * HID_; e += 256) {
    int n = e >> 6, d = e & 63;
    float acc = 0.f;
    if (n < nn) {
#pragma unroll
      for (int p = 0; p < P_; ++p) acc += hb[(p * 32 + n) * STR_ + d];
      acc *= (1.f / 6.f);
    }
    mb[n * STR_ + d] = acc;
  }
  if (wave == 0) __builtin_amdgcn_s_wait_tensorcnt(0);
  __syncthreads();

  // h1 = relu(hm @ set1_W + set1_b)  (32x64 @ 64x64): exactly 8 tiles, one per wave
  {
    int ti = wave >> 2, tc = wave & 3;
    v8f acc = v8f_zero();
#pragma unroll
    for (int kk = 0; kk < HID_; kk += 4) {
      v2f a, bf;
      a.x  = mb[(ti * 16 + lr) * STR_ + kk + 2 * half];
      a.y  = mb[(ti * 16 + lr) * STR_ + kk + 2 * half + 1];
      bf.x = Wa[(kk + 2 * half)     * WSTR_ + tc * 16 + lr];
      bf.y = Wa[(kk + 2 * half + 1) * WSTR_ + tc * 16 + lr];
      acc = wmma4(a, bf, acc);
    }
    int c = tc * 16 + lr;
    float bias = s1b[c];
#pragma unroll
    for (int r = 0; r < 8; ++r) {
      float v = acc[r] + bias;
      tb[(ti * 16 + 8 * half + r) * STR_ + c] = v > 0.f ? v : 0.f;
    }
  }
  __syncthreads();

  // column sum over the 32 node rows
  if (tid < HID_) {
    float acc = 0.f;
    for (int n = 0; n < NM_; ++n) acc += tb[n * STR_ + tid];
    s_col[tid] = acc;
  }
  __syncthreads();

  // h2 = relu(colsum @ set2_W + set2_b); segment-max via bitwise atomicMax (values >= 0)
  if (tid < HID_) {
    float acc = s2b[tid];
    for (int k2 = 0; k2 < HID_; ++k2) acc += s_col[k2] * s2W[k2 * HID_ + tid];
    float v = acc > 0.f ? acc : 0.f;
    atomicMax(&pooled[b * HID_ + tid], __float_as_uint(v));
  }
}

__global__ void idmpnn_out_kernel(const unsigned int* __restrict__ pooled,
                                  const float* __restrict__ outW,  // (HID, 1)
                                  const float* __restrict__ outb,  // (1,)
                                  float* __restrict__ out)         // (B, 1)
{
  int bb = blockIdx.x * blockDim.x + threadIdx.x;
  if (bb < B_) {
    float acc = outb[0];
    for (int d = 0; d < HID_; ++d)
      acc += __uint_as_float(pooled[bb * HID_ + d]) * outW[d];
    out[bb] = acc;
  }
}

extern "C" void kernel_launch(void* const* d_in, const int* in_sizes, int n_in,
                              void* d_out, int out_size, void* d_ws, size_t ws_size,
                              hipStream_t stream) {
  const int*   x        = (const int*)  d_in[0];   // (B,NM,1) int32
  const float* subadj   = (const float*)d_in[1];   // (B,NM,NM)
  const int*   subgs    = (const int*)  d_in[2];   // (S,K) int32
  const int*   num_node = (const int*)  d_in[3];   // (B,) int32
  // d_in[4] = num_subg (unused by the forward pass)
  const float* idemb    = (const float*)d_in[5];
  const float* node_emb = (const float*)d_in[6];
  const float* W1       = (const float*)d_in[7];
  const float* B1       = (const float*)d_in[8];
  const float* W2       = (const float*)d_in[9];
  const float* B2       = (const float*)d_in[10];
  const float* G        = (const float*)d_in[11];
  const float* BE       = (const float*)d_in[12];
  const float* s1W      = (const float*)d_in[13];
  const float* s1b      = (const float*)d_in[14];
  const float* s2W      = (const float*)d_in[15];
  const float* s2b      = (const float*)d_in[16];
  const float* outW     = (const float*)d_in[17];
  const float* outb     = (const float*)d_in[18];

  float* out = (float*)d_out;
  unsigned int* pooled = (unsigned int*)d_ws;  // B*HID uints (16 KB)

  hipLaunchKernelGGL(init_pooled_kernel, dim3((B_ * HID_ + 255) / 256), dim3(256),
                     0, stream, pooled);
  hipLaunchKernelGGL(idmpnn_main_kernel, dim3(S_), dim3(256), 0, stream,
                     x, subadj, subgs, num_node, idemb, node_emb,
                     W1, B1, W2, B2, G, BE, s1W, s1b, s2W, s2b, pooled);
  hipLaunchKernelGGL(idmpnn_out_kernel, dim3(1), dim3(64), 0, stream,
                     pooled, outW, outb, out);
}